// QueryAndGroup_38800734552431
// MI455X (gfx1250) — compile-verified
//
#include <hip/hip_runtime.h>
#include <hip/hip_bf16.h>

// Problem constants (from reference setup_inputs)
#define BATCH   4
#define NPTS    8192
#define NPOINT  2048
#define CFEAT   64
#define NSAMP   32
#define SGRP    33          // NSAMP + 1 (fps_idx prepended)
#define RADIUS2 0.01f       // 0.1^2
#define CH_OUT  70          // 3 + 3 + CFEAT

typedef __attribute__((ext_vector_type(2))) float v2f;
typedef __attribute__((ext_vector_type(8))) float v8f;

// ---------------------------------------------------------------------------
// Kernel 1: ball query via V_WMMA_F32_16X16X4_F32.
//   d2(m,n) = ||c_m||^2 + ||p_n||^2 - 2 c_m . p_n
//   A (16x4)  row m = [-2x_c, -2y_c, -2z_c, 1]
//   B (4x16)  col n = [ x_p ,  y_p ,  z_p , ||p||^2 ]
//   C (16x16) row m = ||c_m||^2      =>  D = d2 tile
//
// C/D layout (ISA 7.12.2): VGPR r -> row M=r in lanes 0-15, row M=r+8 in
// lanes 16-31; column N = lane%16. Selection uses this directly:
// ballot(d[r]<r2) gives per-row hit masks whose bit order == point order,
// so each hitting lane computes its rank via popcount and stores its own
// index -- no serial bit-peeling, no divergent mask-select ladder.
// One wave handles 16 centroids; block = 256 threads = 8 waves = 128 centroids.
// ---------------------------------------------------------------------------
__global__ __launch_bounds__(256) void ball_query_wmma(
    const float* __restrict__ xyz,      // (B,N,3)
    const float* __restrict__ new_xyz,  // (B,NP,3)
    const int*   __restrict__ fps_idx,  // (B,NP)
    int*         __restrict__ idx_out)  // (B*NP, 33)
{
    __shared__ float s_cx[128], s_cy[128], s_cz[128], s_cn2[128];

    const int tid       = threadIdx.x;
    const int cent_blk0 = blockIdx.x * 128;       // global centroid id (b*NP+p)

    if (tid < 128) {
        int g   = cent_blk0 + tid;
        float x = new_xyz[g * 3 + 0];
        float y = new_xyz[g * 3 + 1];
        float z = new_xyz[g * 3 + 2];
        s_cx[tid]  = x; s_cy[tid] = y; s_cz[tid] = z;
        s_cn2[tid] = x * x + y * y + z * z;
    }
    __syncthreads();

    const int  wave   = tid >> 5;
    const int  lane   = tid & 31;
    const int  col    = lane & 15;                // column N within the tile
    const bool hi     = lane >= 16;               // hi half holds rows 8..15
    const int  local0 = wave * 16;                // block-local first centroid
    const int  gcent0 = cent_blk0 + local0;       // global first centroid
    const int  b      = gcent0 / NPOINT;          // NPOINT % 16 == 0 -> uniform
    const float* xb   = xyz + (size_t)b * NPTS * 3;
    const int  rowoff = hi ? 8 : 0;               // row = rowoff + r

    // A matrix (16x4 f32): VGPR0 = K0 (lo) / K2 (hi), VGPR1 = K1 (lo) / K3 (hi)
    v2f amat;
    {
        int m  = local0 + col;
        amat.x = hi ? (-2.0f * s_cz[m]) : (-2.0f * s_cx[m]);
        amat.y = hi ? 1.0f              : (-2.0f * s_cy[m]);
    }
    // C rows: VGPR r -> ||c||^2 of row rowoff + r
    float cn2r[8];
#pragma unroll
    for (int r = 0; r < 8; ++r)
        cn2r[r] = s_cn2[local0 + rowoff + r];

    // Per-row state, replicated across the 16 lanes of each half (all lanes
    // of a half compute identical updates from the wave-uniform ballot).
    int cnt[8], first[8], rowbase[8];
#pragma unroll
    for (int r = 0; r < 8; ++r) {
        cnt[r]     = 0;
        first[r]   = -1;
        rowbase[r] = (gcent0 + rowoff + r) * SGRP;
    }

    const unsigned below = (1u << col) - 1u;      // bits strictly before my column

    for (int j0 = 0; j0 < NPTS; j0 += 16) {
        // B matrix: column N = col; VGPR0 = rows K0/K2, VGPR1 = rows K1/K3
        int   j   = j0 + col;
        float px  = xb[j * 3 + 0];
        float py  = xb[j * 3 + 1];
        float pz  = xb[j * 3 + 2];
        float pn2 = px * px + py * py + pz * pz;
        v2f bmat;
        bmat.x = hi ? pz  : px;
        bmat.y = hi ? pn2 : py;

        v8f cacc;
#pragma unroll
        for (int r = 0; r < 8; ++r) cacc[r] = cn2r[r];

        // 8 args: (neg_a, A, neg_b, B, c_mod, C, reuse_a, reuse_b)
        v8f d = __builtin_amdgcn_wmma_f32_16x16x4_f32(
            false, amat, false, bmat, (short)0, cacc, false, false);

        bool done = true;
#pragma unroll
        for (int r = 0; r < 8; ++r) {
            bool     p    = d[r] < RADIUS2;                    // my (row,col) hit
            unsigned m    = __builtin_amdgcn_ballot_w32(p);    // wave-uniform
            unsigned half = hi ? (m >> 16) : (m & 0xFFFFu);    // my row's 16 bits
            int pos = cnt[r] + __builtin_popcount(half & below);
            if (p && pos < NSAMP)
                idx_out[rowbase[r] + 1 + pos] = j0 + col;      // parallel append
            if (first[r] < 0 && half)
                first[r] = j0 + __builtin_ctz(half);           // earliest member
            cnt[r] += __builtin_popcount(half);
            done = done && (cnt[r] >= NSAMP);
        }
        // Wave-uniform early exit once all 16 rows are full.
        if (__builtin_amdgcn_ballot_w32(done) == 0xFFFFFFFFu) break;
    }

    // Tail: one lane per row (col<8 in each half) pads + writes fps_idx slot.
    if (col < 8) {
        int r = col;
        int c = cnt[r] > NSAMP ? NSAMP : cnt[r];
        int f = first[r] < 0 ? (NPTS - 1) : first[r];
        for (int k = c; k < NSAMP; ++k)
            idx_out[rowbase[r] + 1 + k] = f;                   // pad with first
        idx_out[rowbase[r]] = fps_idx[gcent0 + rowoff + r];    // neighbor 0
    }
}

// ---------------------------------------------------------------------------
// Kernel 2: gather + center. One block per (b, p); 33 indices + centroid
// staged in LDS, 70*33 output elements streamed by 256 threads.
// out[b, c, p, s]: c<3 centered xyz, 3<=c<6 centered xyz (aliased copy),
// c>=6 -> features channel c-6.
// ---------------------------------------------------------------------------
__global__ __launch_bounds__(256) void group_gather(
    const float* __restrict__ xyz,      // (B,N,3)
    const float* __restrict__ new_xyz,  // (B,NP,3)
    const float* __restrict__ feat,     // (B,C,N)
    const int*   __restrict__ idx,      // (B*NP, 33)
    float*       __restrict__ out)      // (B,70,NP,33)
{
    __shared__ int   s_idx[SGRP];
    __shared__ float s_c[3];

    const int g = blockIdx.x;           // 0 .. B*NP-1
    const int b = g / NPOINT;
    const int p = g - b * NPOINT;

    if (threadIdx.x < SGRP) s_idx[threadIdx.x] = idx[g * SGRP + threadIdx.x];
    if (threadIdx.x < 3)    s_c[threadIdx.x]   = new_xyz[g * 3 + threadIdx.x];
    __syncthreads();

    const float* xb = xyz  + (size_t)b * NPTS * 3;
    const float* fb = feat + (size_t)b * CFEAT * NPTS;

    for (int t = threadIdx.x; t < CH_OUT * SGRP; t += blockDim.x) {
        int c = t / SGRP;
        int s = t - c * SGRP;
        int j = s_idx[s];
        float v;
        if (c < 6) {
            int k = c % 3;
            v = xb[(size_t)j * 3 + k] - s_c[k];
        } else {
            v = fb[(size_t)(c - 6) * NPTS + j];
        }
        out[(((size_t)b * CH_OUT + c) * NPOINT + p) * SGRP + s] = v;
    }
}

extern "C" void kernel_launch(void* const* d_in, const int* in_sizes, int n_in,
                              void* d_out, int out_size, void* d_ws, size_t ws_size,
                              hipStream_t stream) {
    const float* xyz     = (const float*)d_in[0];   // (4,8192,3)
    const float* new_xyz = (const float*)d_in[1];   // (4,2048,3)
    const float* feat    = (const float*)d_in[2];   // (4,64,8192)
    const int*   fps     = (const int*)  d_in[3];   // (4,2048)
    float* out   = (float*)d_out;                   // (4,70,2048,33)
    int*   idxws = (int*)d_ws;                      // (4*2048, 33) ints

    // Phase 1: WMMA-based ball query -> index table in workspace.
    ball_query_wmma<<<(BATCH * NPOINT) / 128, 256, 0, stream>>>(
        xyz, new_xyz, fps, idxws);

    // Phase 2: gather + center into the output tensor.
    group_gather<<<BATCH * NPOINT, 256, 0, stream>>>(
        xyz, new_xyz, feat, idxws, out);
}